// MultiscaleDeformableAttention_52089363365898
// MI455X (gfx1250) — compile-verified
//
#include <hip/hip_runtime.h>
#include <math.h>

typedef float v2f __attribute__((ext_vector_type(2)));
typedef float v8f __attribute__((ext_vector_type(8)));

#define NB  8      // batch
#define NQ  1024   // queries
#define NI  21760  // flattened pyramid length
#define NH  8      // heads
#define NL  4      // levels
#define NP  4      // points
#define NCH 32     // channels per head

// ---------------------------------------------------------------------------
// Generic fp32 WMMA GEMM:  C[M,N] = A[M,K] * B[K,N] + bias[N]
// One wave per 64(M) x 64(N) tile; K stepped by 4 via V_WMMA_F32_16X16X4_F32.
// 4 A-frags x 4 B-frags -> 16 WMMA per K-step against 12 loads (4x b64 A,
// 8x b32 B): each B fragment is reused by 4 A fragments.
// A-frag (16x4 f32, ISA 7.12.2): lanes 0-15 -> M=lane, K={0,1}; lanes 16-31 ->
// M=lane-16, K={2,3}.  B-frag mirrors with lanes holding N.  C/D: VGPR r holds
// rows {r, r+8} across the two lane halves.
// ---------------------------------------------------------------------------
__global__ __launch_bounds__(32)
void gemm64x64_f32_wmma(const float* __restrict__ A,
                        const float* __restrict__ B,
                        const float* __restrict__ bias,
                        float* __restrict__ C,
                        int M, int K, int N)
{
    const int lane = threadIdx.x;
    const int half = lane >> 4;      // 0 or 1
    const int l16  = lane & 15;
    const int m0   = blockIdx.x * 64;
    const int n0   = blockIdx.y * 64;

    v8f acc[4][4];
#pragma unroll
    for (int i = 0; i < 4; ++i)
#pragma unroll
        for (int j = 0; j < 4; ++j)
#pragma unroll
            for (int r = 0; r < 8; ++r)
                acc[i][j][r] = 0.0f;

    const float* ap[4];
#pragma unroll
    for (int i = 0; i < 4; ++i)
        ap[i] = A + (size_t)(m0 + i * 16 + l16) * K + 2 * half;

    const float* bp = B + (size_t)(2 * half) * N + n0 + l16;

    for (int k = 0; k < K; k += 4) {
        v2f a[4];
#pragma unroll
        for (int i = 0; i < 4; ++i)
            a[i] = *(const v2f*)(ap[i] + k);

        v2f bf[4];
        const float* bk = bp + (size_t)k * N;
#pragma unroll
        for (int j = 0; j < 4; ++j) {
            bf[j].x = bk[j * 16];
            bf[j].y = bk[j * 16 + N];
        }

#pragma unroll
        for (int i = 0; i < 4; ++i)
#pragma unroll
            for (int j = 0; j < 4; ++j)
                acc[i][j] = __builtin_amdgcn_wmma_f32_16x16x4_f32(
                    false, a[i], false, bf[j], (short)0, acc[i][j], false, false);
    }

#pragma unroll
    for (int j = 0; j < 4; ++j) {
        const int col = n0 + j * 16 + l16;
        const float bv = bias[col];
#pragma unroll
        for (int i = 0; i < 4; ++i) {
#pragma unroll
            for (int r = 0; r < 8; ++r) {
                const int row = m0 + i * 16 + r + 8 * half;
                C[(size_t)row * N + col] = acc[i][j][r] + bv;
            }
        }
    }
}

// ---------------------------------------------------------------------------
// MSDA core: one wave per (b,n,h); lane == channel (CH==32==wave32).
// Lanes 0-15 (mirrored in lanes 16-31) each own one (l,p) sampling point:
// softmax via xor-shuffles, bilinear weights with zeros-padding masking,
// then broadcast weight/index per point and gather 4 coalesced 128B rows.
// ---------------------------------------------------------------------------
__global__ __launch_bounds__(256)
void msda_sample(const float* __restrict__ qp,    // [B*N, H, L*P*3]
                 const float* __restrict__ refp,  // [B*N, 2]
                 const float* __restrict__ imgp,  // [B, I, H*CH]
                 float* __restrict__ out)         // [B*N, H*CH]
{
    const int wave = (blockIdx.x * blockDim.x + threadIdx.x) >> 5;
    const int lane = threadIdx.x & 31;
    const int h    = wave & (NH - 1);
    const int bn   = wave >> 3;        // NH == 8
    const int b    = bn >> 10;         // NQ == 1024
    const int pt   = lane & 15;        // (l,p) point owned by this lane
    const int l    = pt >> 2;

    const float* q = qp + ((size_t)bn * NH + h) * (NL * NP * 3) + pt * 3;
    const float offx  = q[0];
    const float offy  = q[1];
    const float logit = q[2];

    // softmax over the 16 points (duplicated across lane halves)
    float mx = logit;
#pragma unroll
    for (int s = 1; s < 16; s <<= 1) mx = fmaxf(mx, __shfl_xor(mx, s, 32));
    const float e = expf(logit - mx);
    float se = e;
#pragma unroll
    for (int s = 1; s < 16; s <<= 1) se += __shfl_xor(se, s, 32);
    const float attn = e / se;

    const float rx = refp[(size_t)bn * 2 + 0];
    const float ry = refp[(size_t)bn * 2 + 1];

    const int   sz    = 128 >> l;                            // hh == ww per level
    const int   start = (65536 - (65536 >> (2 * l))) / 3;    // {0,16384,20480,21504}
    const float fsz   = (float)sz;

    // reference divides x-offset by shapes[l,0] and y-offset by shapes[l,1]
    const float px = rx + offx / fsz;
    const float py = ry + offy / fsz;
    const float x  = px * fsz - 0.5f;                        // ALIGN_CORNERS=False
    const float y  = py * fsz - 0.5f;
    const float x0f = floorf(x), y0f = floorf(y);
    const float wx = x - x0f, wy = y - y0f;
    const int x0 = (int)x0f, x1 = x0 + 1;
    const int y0 = (int)y0f, y1 = y0 + 1;

    const bool vx0 = (x0 >= 0) & (x0 < sz);
    const bool vx1 = (x1 >= 0) & (x1 < sz);
    const bool vy0 = (y0 >= 0) & (y0 < sz);
    const bool vy1 = (y1 >= 0) & (y1 < sz);

    float w00 = attn * (1.f - wx) * (1.f - wy) * ((vx0 & vy0) ? 1.f : 0.f);
    float w10 = attn * wx         * (1.f - wy) * ((vx1 & vy0) ? 1.f : 0.f);
    float w01 = attn * (1.f - wx) * wy         * ((vx0 & vy1) ? 1.f : 0.f);
    float w11 = attn * wx         * wy         * ((vx1 & vy1) ? 1.f : 0.f);

    const int xc0 = min(max(x0, 0), sz - 1);
    const int xc1 = min(max(x1, 0), sz - 1);
    const int yc0 = min(max(y0, 0), sz - 1);
    const int yc1 = min(max(y1, 0), sz - 1);

    const int i00 = start + yc0 * sz + xc0;
    const int i10 = start + yc0 * sz + xc1;
    const int i01 = start + yc1 * sz + xc0;
    const int i11 = start + yc1 * sz + xc1;

    const float* base = imgp + (size_t)b * NI * (NH * NCH) + h * NCH + lane;

    float acc = 0.f;
#pragma unroll
    for (int p = 0; p < 16; ++p) {
        const float c00 = __shfl(w00, p, 32);
        const float c10 = __shfl(w10, p, 32);
        const float c01 = __shfl(w01, p, 32);
        const float c11 = __shfl(w11, p, 32);
        const int   j00 = __shfl(i00, p, 32);
        const int   j10 = __shfl(i10, p, 32);
        const int   j01 = __shfl(i01, p, 32);
        const int   j11 = __shfl(i11, p, 32);
        acc += c00 * base[(size_t)j00 * (NH * NCH)];
        acc += c10 * base[(size_t)j10 * (NH * NCH)];
        acc += c01 * base[(size_t)j01 * (NH * NCH)];
        acc += c11 * base[(size_t)j11 * (NH * NCH)];
    }
    out[(size_t)bn * (NH * NCH) + h * NCH + lane] = acc;
}

extern "C" void kernel_launch(void* const* d_in, const int* in_sizes, int n_in,
                              void* d_out, int out_size, void* d_ws, size_t ws_size,
                              hipStream_t stream)
{
    const float* img   = (const float*)d_in[0];
    // d_in[1] img_shapes: compile-time constants, unused
    const float* quer  = (const float*)d_in[2];
    const float* refp  = (const float*)d_in[3];
    const float* W_img = (const float*)d_in[4];
    const float* b_img = (const float*)d_in[5];
    const float* W_q   = (const float*)d_in[6];
    const float* b_q   = (const float*)d_in[7];
    const float* W_out = (const float*)d_in[8];
    const float* b_out = (const float*)d_in[9];
    float* out = (float*)d_out;

    char*  ws   = (char*)d_ws;
    float* imgp = (float*)ws;                                            // 8*21760*256 f32 (178 MB)
    float* qpv  = (float*)(ws + (size_t)NB * NI * 256 * sizeof(float));  // 8192*384 f32
    float* msda = qpv + (size_t)NB * NQ * (NH * NL * NP * 3);            // 8192*256 f32

    // 1) qp = queries @ W_q + b_q          [8192 x 384]
    gemm64x64_f32_wmma<<<dim3((NB * NQ) / 64, 384 / 64), 32, 0, stream>>>(
        quer, W_q, b_q, qpv, NB * NQ, 256, 384);
    // 2) img_p = img @ W_img + b_img       [174080 x 256]  (dominant GEMM)
    gemm64x64_f32_wmma<<<dim3((NB * NI) / 64, 256 / 64), 32, 0, stream>>>(
        img, W_img, b_img, imgp, NB * NI, 256, 256);
    // 3) softmax + bilinear sampling + head reduce
    msda_sample<<<(NB * NQ * NH) / 8, 256, 0, stream>>>(qpv, refp, imgp, msda);
    // 4) out = msda @ W_out + b_out        [8192 x 256]
    gemm64x64_f32_wmma<<<dim3((NB * NQ) / 64, 256 / 64), 32, 0, stream>>>(
        msda, W_out, b_out, out, NB * NQ, 256, 256);
}